// TransitionDown_687194767480
// MI455X (gfx1250) — compile-verified
//
#include <hip/hip_runtime.h>

typedef float v2f __attribute__((ext_vector_type(2)));
typedef float v8f __attribute__((ext_vector_type(8)));

#define BB   4
#define NN   8192
#define DD   128
#define KK   2048
#define NSS  16
#define C0   131
#define C0P  132
#define C1   256
#define MTOT 131072   // B*K*NS rows
#define EPSV 1e-5f

// ---------------------------------------------------------------------------
// 1) Farthest point sampling: one block per batch, 1024 threads, 8 pts/thread
// ---------------------------------------------------------------------------
__global__ __launch_bounds__(1024) void fps_kernel(const float* __restrict__ xyz,
                                                   int* __restrict__ fidx,
                                                   float* __restrict__ newxyz) {
  const int b   = blockIdx.x;
  const int tid = threadIdx.x;
  const int lane = tid & 31, wid = tid >> 5;
  const float* X = xyz + (size_t)b * NN * 3;

  float px[8], py[8], pz[8], dist[8];
  int pid[8];
#pragma unroll
  for (int i = 0; i < 8; ++i) {
    int p = i * 1024 + tid;
    pid[i] = p;
    px[i] = X[p * 3 + 0]; py[i] = X[p * 3 + 1]; pz[i] = X[p * 3 + 2];
    dist[i] = 1e10f;
  }

  __shared__ float rv[32];
  __shared__ int   ri[32];
  __shared__ int   far_sh;

  int far = 0;
  for (int k = 0; k < KK; ++k) {
    if (tid == 0) fidx[b * KK + k] = far;
    if (tid < 3) newxyz[((size_t)b * KK + k) * 3 + tid] = X[far * 3 + tid];
    float cx = X[far * 3 + 0], cy = X[far * 3 + 1], cz = X[far * 3 + 2];

    float bv = -1.0f; int bi = 0;
#pragma unroll
    for (int i = 0; i < 8; ++i) {
      float dx = px[i] - cx, dy = py[i] - cy, dz = pz[i] - cz;
      float d  = dx * dx + dy * dy + dz * dz;
      float nd = fminf(dist[i], d);
      dist[i] = nd;
      if (nd > bv) { bv = nd; bi = pid[i]; }
    }
#pragma unroll
    for (int off = 16; off >= 1; off >>= 1) {
      float ov = __shfl_xor(bv, off, 32);
      int   oi = __shfl_xor(bi, off, 32);
      if (ov > bv || (ov == bv && oi < bi)) { bv = ov; bi = oi; }
    }
    if (lane == 0) { rv[wid] = bv; ri[wid] = bi; }
    __syncthreads();
    if (wid == 0) {
      bv = rv[lane]; bi = ri[lane];
#pragma unroll
      for (int off = 16; off >= 1; off >>= 1) {
        float ov = __shfl_xor(bv, off, 32);
        int   oi = __shfl_xor(bi, off, 32);
        if (ov > bv || (ov == bv && oi < bi)) { bv = ov; bi = oi; }
      }
      if (lane == 0) far_sh = bi;
    }
    __syncthreads();
    far = far_sh;
  }
}

// ---------------------------------------------------------------------------
// 2) KNN: one thread per query, LDS-tiled point cloud, register top-16
// ---------------------------------------------------------------------------
#define KNT 1024
__global__ __launch_bounds__(256) void knn_kernel(const float* __restrict__ xyz,
                                                  const float* __restrict__ newxyz,
                                                  int* __restrict__ knn) {
  __shared__ float sp[KNT * 3];
  const int tid = threadIdx.x;
  const int b = blockIdx.x >> 3;
  const int k = ((blockIdx.x & 7) << 8) + tid;
  const size_t qg = (size_t)b * KK + k;
  const float cx = newxyz[qg * 3 + 0];
  const float cy = newxyz[qg * 3 + 1];
  const float cz = newxyz[qg * 3 + 2];
  const float* X = xyz + (size_t)b * NN * 3;

  float dl[16]; int il[16];
#pragma unroll
  for (int u = 0; u < 16; ++u) { dl[u] = 3e38f; il[u] = 0; }
  float vmax = 3e38f; int pmax = 0;

  for (int t = 0; t < NN / KNT; ++t) {
    for (int i = tid; i < KNT * 3; i += 256) sp[i] = X[t * KNT * 3 + i];
    __syncthreads();
    for (int j = 0; j < KNT; ++j) {
      float dx = sp[j * 3 + 0] - cx, dy = sp[j * 3 + 1] - cy, dz = sp[j * 3 + 2] - cz;
      float d = dx * dx + dy * dy + dz * dz;
      if (d < vmax) {
        int id = t * KNT + j;
#pragma unroll
        for (int u = 0; u < 16; ++u) if (u == pmax) { dl[u] = d; il[u] = id; }
        vmax = dl[0]; pmax = 0;
#pragma unroll
        for (int u = 1; u < 16; ++u) if (dl[u] > vmax) { vmax = dl[u]; pmax = u; }
      }
    }
    __syncthreads();
  }
#pragma unroll
  for (int u = 0; u < 16; ++u) knn[qg * NSS + u] = il[u];
}

// ---------------------------------------------------------------------------
// 3) Pad W1 [256x131] -> W1p [256x132] (zero last col) for aligned B loads
// ---------------------------------------------------------------------------
__global__ __launch_bounds__(256) void padw1_kernel(const float* __restrict__ W1,
                                                    float* __restrict__ W1p) {
  int idx = blockIdx.x * 256 + threadIdx.x;      // 256*132 = 33792 total
  int o = idx / C0P, c = idx - o * C0P;
  W1p[idx] = (c < C0) ? W1[o * C0 + c] : 0.0f;
}

// ---------------------------------------------------------------------------
// 4) GEMM1: gather(feat) -> y1 = feat @ W1^T + b1, plus per-block BN partials
//    block = 4 queries (64 rows) x 256 cols; wave = 16 rows x 128 cols
// ---------------------------------------------------------------------------
__global__ __launch_bounds__(256) void gemm1_kernel(const float* __restrict__ xyz,
                                                    const float* __restrict__ points,
                                                    const float* __restrict__ newxyz,
                                                    const int* __restrict__ knn,
                                                    const float* __restrict__ W1p,
                                                    const float* __restrict__ b1,
                                                    float* __restrict__ y1,
                                                    float* __restrict__ part) {
  __shared__ float sA[64 * C0P];
  __shared__ float wsum[4 * 256];
  __shared__ float wsq[4 * 256];
  const int tid = threadIdx.x;
  const int qbase = blockIdx.x * 4;

  for (int e = tid; e < 64 * C0P; e += 256) {
    int row = e / C0P, c = e - row * C0P;
    int qg = qbase + (row >> 4);
    int n  = row & 15;
    int b  = qg >> 11;                                  // / K
    int id = knn[(size_t)qg * NSS + n];
    float v;
    if (c < 3)       v = xyz[((size_t)b * NN + id) * 3 + c] - newxyz[(size_t)qg * 3 + c];
    else if (c < C0) v = points[((size_t)b * NN + id) * DD + (c - 3)];
    else             v = 0.0f;
    sA[e] = v;
  }
  __syncthreads();

  const int lane = tid & 31;
  const int w = tid >> 5;
  const int ql = w & 3;
  const int colhalf = w >> 2;
  const int m16 = lane & 15;
  const int hi = (lane >> 4) & 1;
  const int rowbase = ql * 16;
  const int o0 = colhalf * 128;

  v8f acc[8];
#pragma unroll
  for (int t = 0; t < 8; ++t) acc[t] = (v8f)0.0f;

  for (int s = 0; s < 33; ++s) {
    int c = s * 4 + hi * 2;
    v2f a = *(const v2f*)&sA[(rowbase + m16) * C0P + c];
#pragma unroll
    for (int t = 0; t < 8; ++t) {
      int o = o0 + t * 16 + m16;
      v2f bf = *(const v2f*)&W1p[o * C0P + c];
      acc[t] = __builtin_amdgcn_wmma_f32_16x16x4_f32(false, a, false, bf, (short)0,
                                                     acc[t], false, false);
    }
  }

#pragma unroll
  for (int t = 0; t < 8; ++t) {
    int o = o0 + t * 16 + m16;
    float bias = b1[o];
    float ss = 0.0f, sq = 0.0f;
#pragma unroll
    for (int v = 0; v < 8; ++v) {
      float y = acc[t][v] + bias;
      int m = hi ? (v + 8) : v;
      y1[((size_t)blockIdx.x * 64 + rowbase + m) * C1 + o] = y;
      ss += y; sq += y * y;
    }
    ss += __shfl_xor(ss, 16, 32);
    sq += __shfl_xor(sq, 16, 32);
    if (!hi) { wsum[ql * 256 + o] = ss; wsq[ql * 256 + o] = sq; }
  }
  __syncthreads();
  float ps = 0.0f, pq = 0.0f;
#pragma unroll
  for (int qi = 0; qi < 4; ++qi) { ps += wsum[qi * 256 + tid]; pq += wsq[qi * 256 + tid]; }
  part[(size_t)blockIdx.x * 512 + tid]       = ps;
  part[(size_t)blockIdx.x * 512 + 256 + tid] = pq;
}

// ---------------------------------------------------------------------------
// 5) Finalize BN stats: reduce block partials -> affine (a, c) per channel
// ---------------------------------------------------------------------------
__global__ __launch_bounds__(256) void finalize_kernel(const float* __restrict__ part,
                                                       int nblocks,
                                                       const float* __restrict__ g,
                                                       const float* __restrict__ be,
                                                       float* __restrict__ aOut,
                                                       float* __restrict__ cOut) {
  int ch = threadIdx.x;
  float s = 0.0f, s2 = 0.0f;
  for (int i = 0; i < nblocks; ++i) {
    s  += part[(size_t)i * 512 + ch];
    s2 += part[(size_t)i * 512 + 256 + ch];
  }
  const float M = (float)MTOT;
  float mean = s / M;
  float var  = s2 / M - mean * mean;
  float inv  = rsqrtf(var + EPSV);
  float a = g[ch] * inv;
  aOut[ch] = a;
  cOut[ch] = be[ch] - mean * a;
}

// ---------------------------------------------------------------------------
// 6) GEMM2: z = relu(a1*y1+c1); y2 = z @ W2^T + b2. Emits per-(query,chan)
//    max/min over 16 neighbors + BN partials. No y2 store.
// ---------------------------------------------------------------------------
__global__ __launch_bounds__(256) void gemm2_kernel(const float* __restrict__ y1,
                                                    const float* __restrict__ W2,
                                                    const float* __restrict__ b2,
                                                    const float* __restrict__ a1,
                                                    const float* __restrict__ c1,
                                                    float* __restrict__ ymax,
                                                    float* __restrict__ ymin,
                                                    float* __restrict__ part) {
  __shared__ float sZ[64 * C0P];     // use cols [0,128) per half, stride 132 (pad)
  __shared__ float sac[512];
  __shared__ float wsum[4 * 256];
  __shared__ float wsq[4 * 256];
  const int tid = threadIdx.x;
  sac[tid] = a1[tid];
  sac[256 + tid] = c1[tid];

  const int lane = tid & 31;
  const int w = tid >> 5;
  const int ql = w & 3;
  const int colhalf = w >> 2;
  const int m16 = lane & 15;
  const int hi = (lane >> 4) & 1;
  const int rowbase = ql * 16;
  const int o0 = colhalf * 128;
  const size_t R0 = (size_t)blockIdx.x * 64;

  v8f acc[8];
#pragma unroll
  for (int t = 0; t < 8; ++t) acc[t] = (v8f)0.0f;

  for (int h = 0; h < 2; ++h) {
    __syncthreads();
    for (int e = tid; e < 64 * 128; e += 256) {
      int row = e >> 7, cc = e & 127;
      int ch = h * 128 + cc;
      float v = y1[(R0 + row) * C1 + ch];
      sZ[row * C0P + cc] = fmaxf(v * sac[ch] + sac[256 + ch], 0.0f);
    }
    __syncthreads();
    for (int s = 0; s < 32; ++s) {
      int cl = s * 4 + hi * 2;
      int c  = h * 128 + cl;
      v2f a = *(const v2f*)&sZ[(rowbase + m16) * C0P + cl];
#pragma unroll
      for (int t = 0; t < 8; ++t) {
        int o = o0 + t * 16 + m16;
        v2f bf = *(const v2f*)&W2[o * C1 + c];
        acc[t] = __builtin_amdgcn_wmma_f32_16x16x4_f32(false, a, false, bf, (short)0,
                                                       acc[t], false, false);
      }
    }
  }

  const size_t qg = (size_t)blockIdx.x * 4 + ql;
#pragma unroll
  for (int t = 0; t < 8; ++t) {
    int o = o0 + t * 16 + m16;
    float bias = b2[o];
    float mx = -3e38f, mn = 3e38f, ss = 0.0f, sq = 0.0f;
#pragma unroll
    for (int v = 0; v < 8; ++v) {
      float y = acc[t][v] + bias;
      mx = fmaxf(mx, y); mn = fminf(mn, y);
      ss += y; sq += y * y;
    }
    mx = fmaxf(mx, __shfl_xor(mx, 16, 32));
    mn = fminf(mn, __shfl_xor(mn, 16, 32));
    ss += __shfl_xor(ss, 16, 32);
    sq += __shfl_xor(sq, 16, 32);
    if (!hi) {
      ymax[qg * C1 + o] = mx;
      ymin[qg * C1 + o] = mn;
      wsum[ql * 256 + o] = ss;
      wsq[ql * 256 + o]  = sq;
    }
  }
  __syncthreads();
  float ps = 0.0f, pq = 0.0f;
#pragma unroll
  for (int qi = 0; qi < 4; ++qi) { ps += wsum[qi * 256 + tid]; pq += wsq[qi * 256 + tid]; }
  part[(size_t)blockIdx.x * 512 + tid]       = ps;
  part[(size_t)blockIdx.x * 512 + 256 + tid] = pq;
}

// ---------------------------------------------------------------------------
// 7) Final: new_points = relu(a2*(a2>=0 ? max : min) + c2)  (BN+relu+maxpool)
// ---------------------------------------------------------------------------
__global__ __launch_bounds__(256) void maxout_kernel(const float* __restrict__ ymax,
                                                     const float* __restrict__ ymin,
                                                     const float* __restrict__ a2,
                                                     const float* __restrict__ c2,
                                                     float* __restrict__ out) {
  int g = blockIdx.x * 256 + threadIdx.x;
  int o = g & 255;
  float a = a2[o], c = c2[o];
  float v = (a >= 0.0f) ? ymax[g] : ymin[g];
  out[g] = fmaxf(a * v + c, 0.0f);
}

// ---------------------------------------------------------------------------
extern "C" void kernel_launch(void* const* d_in, const int* in_sizes, int n_in,
                              void* d_out, int out_size, void* d_ws, size_t ws_size,
                              hipStream_t stream) {
  const float* xyz    = (const float*)d_in[0];
  const float* points = (const float*)d_in[1];
  const float* W1     = (const float*)d_in[2];
  const float* b1     = (const float*)d_in[3];
  const float* g1     = (const float*)d_in[4];
  const float* be1    = (const float*)d_in[5];
  const float* W2     = (const float*)d_in[6];
  const float* b2     = (const float*)d_in[7];
  const float* g2     = (const float*)d_in[8];
  const float* be2    = (const float*)d_in[9];

  float* newxyz = (float*)d_out;                      // [B,K,3]
  float* newpts = (float*)d_out + (size_t)BB * KK * 3; // [B,K,256]

  char* base = (char*)d_ws;
  size_t off = 0;
  auto alloc = [&](size_t bytes) {
    size_t o = off;
    off = (off + bytes + 255) & ~(size_t)255;
    return o;
  };
  int*   fidx  = (int*)(base + alloc((size_t)BB * KK * 4));
  int*   knn   = (int*)(base + alloc((size_t)BB * KK * NSS * 4));
  float* W1p   = (float*)(base + alloc((size_t)C1 * C0P * 4));
  float* y1    = (float*)(base + alloc((size_t)MTOT * C1 * 4));
  float* ymax  = (float*)(base + alloc((size_t)BB * KK * C1 * 4));
  float* ymin  = (float*)(base + alloc((size_t)BB * KK * C1 * 4));
  float* part1 = (float*)(base + alloc((size_t)2048 * 512 * 4));
  float* part2 = (float*)(base + alloc((size_t)2048 * 512 * 4));
  float* a1    = (float*)(base + alloc(256 * 4));
  float* c1    = (float*)(base + alloc(256 * 4));
  float* a2    = (float*)(base + alloc(256 * 4));
  float* c2    = (float*)(base + alloc(256 * 4));

  hipLaunchKernelGGL(padw1_kernel, dim3(C0P), dim3(256), 0, stream, W1, W1p);
  hipLaunchKernelGGL(fps_kernel, dim3(BB), dim3(1024), 0, stream, xyz, fidx, newxyz);
  hipLaunchKernelGGL(knn_kernel, dim3(BB * KK / 256), dim3(256), 0, stream, xyz, newxyz, knn);
  hipLaunchKernelGGL(gemm1_kernel, dim3(BB * KK / 4), dim3(256), 0, stream,
                     xyz, points, newxyz, knn, W1p, b1, y1, part1);
  hipLaunchKernelGGL(finalize_kernel, dim3(1), dim3(256), 0, stream, part1, 2048, g1, be1, a1, c1);
  hipLaunchKernelGGL(gemm2_kernel, dim3(BB * KK / 4), dim3(256), 0, stream,
                     y1, W2, b2, a1, c1, ymax, ymin, part2);
  hipLaunchKernelGGL(finalize_kernel, dim3(1), dim3(256), 0, stream, part2, 2048, g2, be2, a2, c2);
  hipLaunchKernelGGL(maxout_kernel, dim3(BB * KK), dim3(256), 0, stream, ymax, ymin, a2, c2, newpts);
  (void)in_sizes; (void)n_in; (void)out_size; (void)ws_size;
}